// PCGATLayer_24696061952076
// MI455X (gfx1250) — compile-verified
//
#include <hip/hip_runtime.h>
#include <hip/hip_bf16.h>

typedef float v2f __attribute__((ext_vector_type(2)));
typedef float v8f __attribute__((ext_vector_type(8)));

#define IN_F   64
#define OUT_F  32
#define NSTEPS 20
#define LR_C   0.1f
#define SLOPE_C 0.2f

// ---------- monotone float<->uint encoding for atomic segment-max ----------
__device__ __forceinline__ unsigned enc_f32(float f) {
    unsigned u = __float_as_uint(f);
    return (u & 0x80000000u) ? ~u : (u | 0x80000000u);
}
__device__ __forceinline__ float dec_f32(unsigned u) {
    return (u & 0x80000000u) ? __uint_as_float(u & 0x7FFFFFFFu)
                             : __uint_as_float(~u);
}

// ---------- f32 WMMA GEMM: T[N,32] = M[N,64] @ W[32,64]^T ----------
// 4 waves per block; wave w -> row tile (w>>1), col tile (w&1).
__global__ void gemm_wmma_kernel(const float* __restrict__ M,
                                 const float* __restrict__ W,
                                 float* __restrict__ T, int N) {
    const int wave = threadIdx.x >> 5;
    const int lane = threadIdx.x & 31;
    const int half = lane >> 4;      // 0: lanes 0-15, 1: lanes 16-31
    const int l16  = lane & 15;
    const int rowBase = blockIdx.x * 32 + (wave >> 1) * 16;
    const int colBase = (wave & 1) * 16;
    const int nCol = colBase + l16;  // 0..31 (always valid: OUT_F==32)

    v8f c = {};
#pragma unroll
    for (int k0 = 0; k0 < IN_F; k0 += 4) {
        const int kb = k0 + 2 * half;
        int r = rowBase + l16;
        int rc = (r < N) ? r : (N - 1);          // clamp (store is masked later)
        v2f av = *(const v2f*)(M + (size_t)rc * IN_F + kb);
        v2f bv = *(const v2f*)(W + (size_t)nCol * IN_F + kb);
        c = __builtin_amdgcn_wmma_f32_16x16x4_f32(
                false, av, false, bv, (short)0, c, false, false);
    }
#pragma unroll
    for (int r = 0; r < 8; ++r) {
        int row = rowBase + r + 8 * half;
        if (row < N) T[(size_t)row * OUT_F + nCol] = c[r];
    }
}

// ---------- simple zero fill ----------
__global__ void zero_f32_kernel(float* __restrict__ p, int n) {
    int i = blockIdx.x * 256 + threadIdx.x;
    if (i < n) p[i] = 0.0f;
}

// zero agg (N*32), smax (N, encoded -inf == 0), ssum (N)
__global__ void zero_step_kernel(float* __restrict__ agg,
                                 unsigned* __restrict__ smax,
                                 float* __restrict__ ssum, int N) {
    int i = blockIdx.x * 256 + threadIdx.x;
    if (i < N * OUT_F) agg[i] = 0.0f;
    if (i < N) { smax[i] = 0u; ssum[i] = 0.0f; }
}

// ---------- degree + alpha0 ----------
__global__ void deg_kernel(const int* __restrict__ dstI,
                           float* __restrict__ deg, int E) {
    int e = blockIdx.x * 256 + threadIdx.x;
    if (e < E) atomicAdd(deg + dstI[e], 1.0f);
}
__global__ void alpha0_kernel(const int* __restrict__ dstI,
                              const float* __restrict__ deg,
                              float* __restrict__ alpha, int E) {
    int e = blockIdx.x * 256 + threadIdx.x;
    if (e < E) alpha[e] = 1.0f / (deg[dstI[e]] + 1e-8f);
}

// ---------- scatter: agg[dst] += alpha[e] * transformed[src]; 1 wave/edge ----------
__global__ void scatter_kernel(const int* __restrict__ srcI,
                               const int* __restrict__ dstI,
                               const float* __restrict__ alpha,
                               const float* __restrict__ transformed,
                               float* __restrict__ agg, int E) {
    int e = blockIdx.x * 8 + (threadIdx.x >> 5);
    int f = threadIdx.x & 31;
    if (e >= E) return;                         // wave-uniform branch
    float al = alpha[e];
    int s = srcI[e], d = dstI[e];
    atomicAdd(agg + (size_t)d * OUT_F + f, al * transformed[(size_t)s * OUT_F + f]);
}

// ---------- per-node: mu_hat=relu(agg); err=mu-mu_hat; dots; mu update ----------
// mode 0: write s_src/s_dst, mu -= LR*err.  mode 1: write errors only.
__global__ void node_kernel(float* __restrict__ mu,
                            const float* __restrict__ agg,
                            const float* __restrict__ a,
                            float* __restrict__ s_src,
                            float* __restrict__ s_dst,
                            float* __restrict__ errors,
                            int N, int mode) {
    int node = blockIdx.x * 8 + (threadIdx.x >> 5);
    int f = threadIdx.x & 31;
    if (node >= N) return;                      // wave-uniform branch
    float g = agg[(size_t)node * OUT_F + f];
    float mh = g > 0.0f ? g : 0.0f;
    float m = mu[(size_t)node * OUT_F + f];
    float err = m - mh;
    if (mode == 0) {
        float vs = err * a[f];
        float vd = err * a[OUT_F + f];
#pragma unroll
        for (int off = 16; off; off >>= 1) {
            vs += __shfl_xor(vs, off, 32);
            vd += __shfl_xor(vd, off, 32);
        }
        if (f == 0) { s_src[node] = vs; s_dst[node] = vd; }
        mu[(size_t)node * OUT_F + f] = m - LR_C * err;
    } else {
        errors[(size_t)node * OUT_F + f] = err;
    }
}

// ---------- per-edge score + segment max ----------
__global__ void score_kernel(const float* __restrict__ s_src,
                             const float* __restrict__ s_dst,
                             const int* __restrict__ srcI,
                             const int* __restrict__ dstI,
                             float* __restrict__ sc,
                             unsigned* __restrict__ smax, int E) {
    int e = blockIdx.x * 256 + threadIdx.x;
    if (e >= E) return;
    float s = s_src[srcI[e]] + s_dst[dstI[e]];
    s = s > 0.0f ? s : SLOPE_C * s;             // leaky relu
    sc[e] = s;
    atomicMax(smax + dstI[e], enc_f32(s));
}

// ---------- per-edge exp + segment sum ----------
__global__ void expsum_kernel(const float* __restrict__ sc,
                              const unsigned* __restrict__ smax,
                              const int* __restrict__ dstI,
                              float* __restrict__ ex,
                              float* __restrict__ ssum, int E) {
    int e = blockIdx.x * 256 + threadIdx.x;
    if (e >= E) return;
    int d = dstI[e];
    float v = expf(sc[e] - dec_f32(smax[d]));
    ex[e] = v;
    atomicAdd(ssum + d, v);
}

// ---------- per-edge normalize ----------
__global__ void norm_kernel(const float* __restrict__ ex,
                            const float* __restrict__ ssum,
                            const int* __restrict__ dstI,
                            float* __restrict__ alpha, int E) {
    int e = blockIdx.x * 256 + threadIdx.x;
    if (e >= E) return;
    alpha[e] = ex[e] / (ssum[dstI[e]] + 1e-8f);
}

extern "C" void kernel_launch(void* const* d_in, const int* in_sizes, int n_in,
                              void* d_out, int out_size, void* d_ws, size_t ws_size,
                              hipStream_t stream) {
    const float* mu_upper = (const float*)d_in[0];
    const float* W        = (const float*)d_in[1];
    const float* a        = (const float*)d_in[2];
    const int*   ei       = (const int*)d_in[3];

    const int N = in_sizes[0] / IN_F;
    const int E = in_sizes[3] / 2;
    const int* srcI = ei;
    const int* dstI = ei + E;

    // outputs: mu (N*32) | errors (N*32) | alpha (E)
    float* mu     = (float*)d_out;
    float* errors = mu + (size_t)N * OUT_F;
    float* alpha  = errors + (size_t)N * OUT_F;

    // workspace carve (256B aligned)
    char* w = (char*)d_ws;
    auto carve = [&](size_t bytes) -> void* {
        void* p = (void*)w;
        w += (bytes + 255) & ~(size_t)255;
        return p;
    };
    float*    transformed = (float*)carve((size_t)N * OUT_F * 4);
    float*    agg         = (float*)carve((size_t)N * OUT_F * 4);
    float*    deg         = (float*)carve((size_t)N * 4);
    float*    ssum        = (float*)carve((size_t)N * 4);
    float*    s_src       = (float*)carve((size_t)N * 4);
    float*    s_dst       = (float*)carve((size_t)N * 4);
    unsigned* smax        = (unsigned*)carve((size_t)N * 4);
    float*    sc          = (float*)carve((size_t)E * 4);
    float*    ex          = (float*)carve((size_t)E * 4);

    const dim3 b256(256);
    const int gNF   = (N * OUT_F + 255) / 256;     // per node-feature
    const int gN1   = (N + 255) / 256;             // per node (flat)
    const int gNw   = (N + 7) / 8;                 // 1 wave / node
    const int gE1   = (E + 255) / 256;             // per edge (flat)
    const int gEw   = (E + 7) / 8;                 // 1 wave / edge
    const int gGemm = (N + 31) / 32;

    // init: mu = 0, deg = 0
    zero_f32_kernel<<<gNF, b256, 0, stream>>>(mu, N * OUT_F);
    zero_f32_kernel<<<gN1, b256, 0, stream>>>(deg, N);
    deg_kernel<<<gE1, b256, 0, stream>>>(dstI, deg, E);
    alpha0_kernel<<<gE1, b256, 0, stream>>>(dstI, deg, alpha, E);

    // transformed = mu_upper @ W^T  (f32 WMMA)
    gemm_wmma_kernel<<<gGemm, dim3(128), 0, stream>>>(mu_upper, W, transformed, N);

    for (int t = 0; t < NSTEPS; ++t) {
        zero_step_kernel<<<gNF, b256, 0, stream>>>(agg, smax, ssum, N);
        scatter_kernel<<<gEw, b256, 0, stream>>>(srcI, dstI, alpha, transformed, agg, E);
        node_kernel<<<gNw, b256, 0, stream>>>(mu, agg, a, s_src, s_dst, errors, N, 0);
        score_kernel<<<gE1, b256, 0, stream>>>(s_src, s_dst, srcI, dstI, sc, smax, E);
        expsum_kernel<<<gE1, b256, 0, stream>>>(sc, smax, dstI, ex, ssum, E);
        norm_kernel<<<gE1, b256, 0, stream>>>(ex, ssum, dstI, alpha, E);
    }

    // final top_down with final alpha; errors = mu - relu(agg)
    zero_step_kernel<<<gNF, b256, 0, stream>>>(agg, smax, ssum, N);
    scatter_kernel<<<gEw, b256, 0, stream>>>(srcI, dstI, alpha, transformed, agg, E);
    node_kernel<<<gNw, b256, 0, stream>>>(mu, agg, a, s_src, s_dst, errors, N, 1);
}